// ContextSR_30691836297675
// MI455X (gfx1250) — compile-verified
//
#include <hip/hip_runtime.h>
#include <hip/hip_bf16.h>

// ---------------- problem constants ----------------
#define Gc      32                  // context channels
#define CE      32                  // extra channels
#define CIN     64                  // G + CE
#define Hh      192
#define Ww      192
#define PS      (Hh * Ww)           // 36864 pixels / plane
#define NB      2                   // feature batches
#define NBB     6                   // effective batches (repeat 3x)
#define C1OUT   68
#define C1PAD   80                  // padded to 5 WMMA M-tiles
#define COUT    260                 // 68 + 3*64
#define TILES   (PS / 16)           // 2304 16-pixel tiles per plane
#define SW      208                 // padded integral-image row stride (64B aligned)
#define SPLANE  (193 * SW)          // 40144 floats per integral plane

typedef __attribute__((ext_vector_type(2))) float v2f;
typedef __attribute__((ext_vector_type(8))) float v8f;

// D = A(16x4) * B(4x16) + C, fp32, full precision
__device__ __forceinline__ v8f wmma_f32_4(v2f a, v2f b, v8f c) {
  return __builtin_amdgcn_wmma_f32_16x16x4_f32(false, a, false, b, (short)0, c,
                                               false, false);
}

// =====================================================================
// Kernel 1: img_ctx = conv1x1(feature, w_ctx) + b_ctx   (only NB batches)
// M = 32 out channels (2 tiles), K = 32, N = 16 pixels per wave.
// =====================================================================
__global__ void k_ctx_conv(const float* __restrict__ feat,
                           const float* __restrict__ w_ctx,
                           const float* __restrict__ b_ctx,
                           float* __restrict__ xctx) {
  __shared__ float lw[Gc * Gc];
  __shared__ float lb[Gc];
  const int tid = threadIdx.x;
  for (int i = tid; i < Gc * Gc; i += 256) lw[i] = w_ctx[i];
  for (int i = tid; i < Gc; i += 256) lb[i] = b_ctx[i];
  __syncthreads();

  const int wave = tid >> 5, lane = tid & 31;
  const int gw = blockIdx.x * 8 + wave;          // 0 .. NB*TILES-1
  const int fb = gw / TILES;
  const int p0 = (gw % TILES) * 16;
  const int n = lane & 15, hf = lane >> 4;       // pixel-in-tile, K-half
  const int m = lane & 15;                       // A-row

  const float* fbase = feat + (size_t)fb * Gc * PS + p0 + n;

  v8f acc0, acc1;
#pragma unroll
  for (int r = 0; r < 8; ++r) {
    acc0[r] = lb[r + 8 * hf];
    acc1[r] = lb[16 + r + 8 * hf];
  }

#pragma unroll
  for (int c0 = 0; c0 < Gc; c0 += 4) {
    const int ka = c0 + 2 * hf;
    v2f b;
    b[0] = fbase[(size_t)ka * PS];
    b[1] = fbase[(size_t)(ka + 1) * PS];
    v2f a0, a1;
    a0[0] = lw[m * Gc + ka];        a0[1] = lw[m * Gc + ka + 1];
    a1[0] = lw[(16 + m) * Gc + ka]; a1[1] = lw[(16 + m) * Gc + ka + 1];
    acc0 = wmma_f32_4(a0, b, acc0);
    acc1 = wmma_f32_4(a1, b, acc1);
  }

  float* ob = xctx + (size_t)fb * Gc * PS + p0 + n;
#pragma unroll
  for (int r = 0; r < 8; ++r) {
    ob[(size_t)(r + 8 * hf) * PS] = acc0[r];
    ob[(size_t)(16 + r + 8 * hf) * PS] = acc1[r];
  }
}

// =====================================================================
// Kernel 2a: per-plane row prefix sums of x and x*x into padded S1,S2.
// One block per (bb, c) plane; one wave per row; lane-parallel scan.
// =====================================================================
__global__ void k_int_rows(const float* __restrict__ xctx,
                           const float* __restrict__ extra,
                           float* __restrict__ S1, float* __restrict__ S2) {
  const int pl = blockIdx.x;                 // 0 .. 6*64-1
  const int bb = pl / CIN, c = pl % CIN;
  const float* src = (c < Gc)
      ? xctx + ((size_t)(bb / 3) * Gc + c) * PS
      : extra + ((size_t)bb * CE + (c - Gc)) * PS;
  float* S1p = S1 + (size_t)pl * SPLANE;
  float* S2p = S2 + (size_t)pl * SPLANE;

  const int tid = threadIdx.x;
  for (int i = tid; i <= Ww; i += 256) { S1p[i] = 0.f; S2p[i] = 0.f; }  // row 0

  const int wave = tid >> 5, lane = tid & 31;
  for (int y = wave; y < Hh; y += 8) {
    const float* row = src + (size_t)y * Ww;
    float* r1 = S1p + (size_t)(y + 1) * SW;
    float* r2 = S2p + (size_t)(y + 1) * SW;
    if (lane == 0) { r1[0] = 0.f; r2[0] = 0.f; }   // col 0
    float c1 = 0.f, c2 = 0.f;
#pragma unroll
    for (int cx = 0; cx < Ww / 32; ++cx) {
      const int x = cx * 32 + lane;
      float v = row[x];
      float v2 = v * v;
      for (int d = 1; d < 32; d <<= 1) {           // inclusive wave scan
        float t1 = __shfl_up(v, d, 32);
        float t2 = __shfl_up(v2, d, 32);
        if (lane >= d) { v += t1; v2 += t2; }
      }
      r1[1 + x] = v + c1;
      r2[1 + x] = v2 + c2;
      c1 += __shfl(v, 31, 32);
      c2 += __shfl(v2, 31, 32);
    }
  }
}

// =====================================================================
// Kernel 2b: in-place column prefix sums on S1,S2 (one thread per column)
// =====================================================================
__global__ void k_int_cols(float* __restrict__ S1, float* __restrict__ S2) {
  const int pl = blockIdx.x;
  const int col = threadIdx.x + 1;                 // block = 192 threads
  float* S1p = S1 + (size_t)pl * SPLANE;
  float* S2p = S2 + (size_t)pl * SPLANE;
  float a1 = 0.f, a2 = 0.f;
  for (int y = 1; y <= Hh; ++y) {
    const size_t idx = (size_t)y * SW + col;
    a1 += S1p[idx]; S1p[idx] = a1;
    a2 += S2p[idx]; S2p[idx] = a2;
  }
}

// =====================================================================
// Kernel 3: out0 = conv1x1(x, w1) + b1  (K=64, O=68 padded to 80 = 5 tiles)
// x is virtual: channels 0..31 from xctx[bb/3], 32..63 from extra[bb].
// =====================================================================
__global__ void k_conv1(const float* __restrict__ xctx,
                        const float* __restrict__ extra,
                        const float* __restrict__ w1,
                        const float* __restrict__ b1,
                        float* __restrict__ out) {
  __shared__ float lw[C1PAD * CIN];
  __shared__ float lb[C1PAD];
  const int tid = threadIdx.x;
  for (int i = tid; i < C1PAD * CIN; i += 256) {
    const int o = i >> 6, cc = i & 63;
    lw[i] = (o < C1OUT) ? w1[o * CIN + cc] : 0.f;
  }
  for (int i = tid; i < C1PAD; i += 256) lb[i] = (i < C1OUT) ? b1[i] : 0.f;
  __syncthreads();

  const int wave = tid >> 5, lane = tid & 31;
  const int gw = blockIdx.x * 8 + wave;            // 0 .. NBB*TILES-1
  const int bb = gw / TILES;
  const int p0 = (gw % TILES) * 16;
  const int fb = bb / 3;
  const int n = lane & 15, hf = lane >> 4, m = lane & 15;

  const float* xc = xctx + (size_t)fb * Gc * PS + p0 + n;
  const float* ex = extra + (size_t)bb * CE * PS + p0 + n;

  v8f acc[5];
#pragma unroll
  for (int mt = 0; mt < 5; ++mt)
#pragma unroll
    for (int r = 0; r < 8; ++r) acc[mt][r] = lb[mt * 16 + r + 8 * hf];

#pragma unroll 4
  for (int c0 = 0; c0 < CIN; c0 += 4) {
    const float* bp = (c0 < Gc) ? (xc + (size_t)c0 * PS)
                                : (ex + (size_t)(c0 - Gc) * PS);
    v2f b;
    b[0] = bp[(size_t)(2 * hf) * PS];
    b[1] = bp[(size_t)(2 * hf + 1) * PS];
#pragma unroll
    for (int mt = 0; mt < 5; ++mt) {
      v2f a;
      a[0] = lw[(mt * 16 + m) * CIN + c0 + 2 * hf];
      a[1] = lw[(mt * 16 + m) * CIN + c0 + 2 * hf + 1];
      acc[mt] = wmma_f32_4(a, b, acc[mt]);
    }
  }

  float* ob = out + (size_t)bb * COUT * PS + p0 + n;
#pragma unroll
  for (int mt = 0; mt < 5; ++mt)
#pragma unroll
    for (int r = 0; r < 8; ++r) {
      const int ch = mt * 16 + r + 8 * hf;
      if (ch < C1OUT) ob[(size_t)ch * PS] = acc[mt][r];
    }
}

// =====================================================================
// Kernel 4: fused box mean/var (from L2-resident S1,S2) + conv1x1(128->64)
// K = 128: for each 4-channel step do one WMMA with mean columns of W and
// one with var columns. O = 64 -> 4 M-tiles. One 16-pixel tile per wave.
// =====================================================================
__device__ __forceinline__ void box_mv(const float* __restrict__ P1,
                                       const float* __restrict__ P2,
                                       int iy0, int iy1, int xlo, int xhi,
                                       float rarea, float& mean, float& var) {
  const float s1 = P1[iy1 + xhi] - P1[iy1 + xlo] - P1[iy0 + xhi] + P1[iy0 + xlo];
  const float s2 = P2[iy1 + xhi] - P2[iy1 + xlo] - P2[iy0 + xhi] + P2[iy0 + xlo];
  mean = s1 * rarea;
  var = s2 * rarea - mean * mean;
}

__global__ void k_scale_conv(const float* __restrict__ S1,
                             const float* __restrict__ S2,
                             const float* __restrict__ wsc,
                             const float* __restrict__ bsc,
                             float* __restrict__ out,
                             int halfw, int chbase) {
  __shared__ float lw[64 * 128];
  __shared__ float lb[64];
  const int tid = threadIdx.x;
  for (int i = tid; i < 64 * 128; i += 256) lw[i] = wsc[i];
  for (int i = tid; i < 64; i += 256) lb[i] = bsc[i];
  __syncthreads();

  const int wave = tid >> 5, lane = tid & 31;
  const int gw = blockIdx.x * 8 + wave;            // 0 .. NBB*TILES-1
  const int bb = gw / TILES;
  const int p0 = (gw % TILES) * 16;
  const int y = p0 / Ww, x0 = p0 - y * Ww;
  const int n = lane & 15, hf = lane >> 4, m = lane & 15;

  const int px = x0 + n;
  const int y0 = max(y - halfw, 0), y1 = min(y + 1 + halfw, Hh);
  const int xlo = max(px - halfw, 0), xhi = min(px + 1 + halfw, Ww);
  const int iy0 = y0 * SW, iy1 = y1 * SW;
  const float rarea = 1.0f / (float)((y1 - y0) * (xhi - xlo));

  const float* S1b = S1 + (size_t)bb * CIN * SPLANE;
  const float* S2b = S2 + (size_t)bb * CIN * SPLANE;

  v8f acc[4];
#pragma unroll
  for (int mt = 0; mt < 4; ++mt)
#pragma unroll
    for (int r = 0; r < 8; ++r) acc[mt][r] = lb[mt * 16 + r + 8 * hf];

  for (int c0 = 0; c0 < CIN; c0 += 4) {
    const int ca = c0 + 2 * hf, cb = ca + 1;
    float m0, v0, m1, v1;
    box_mv(S1b + (size_t)ca * SPLANE, S2b + (size_t)ca * SPLANE,
           iy0, iy1, xlo, xhi, rarea, m0, v0);
    box_mv(S1b + (size_t)cb * SPLANE, S2b + (size_t)cb * SPLANE,
           iy0, iy1, xlo, xhi, rarea, m1, v1);
    v2f bm, bv;
    bm[0] = m0; bm[1] = m1;
    bv[0] = v0; bv[1] = v1;
#pragma unroll
    for (int mt = 0; mt < 4; ++mt) {
      v2f am, av;
      am[0] = lw[(mt * 16 + m) * 128 + ca];
      am[1] = lw[(mt * 16 + m) * 128 + ca + 1];
      av[0] = lw[(mt * 16 + m) * 128 + 64 + ca];
      av[1] = lw[(mt * 16 + m) * 128 + 64 + ca + 1];
      acc[mt] = wmma_f32_4(am, bm, acc[mt]);
      acc[mt] = wmma_f32_4(av, bv, acc[mt]);
    }
  }

  float* ob = out + (size_t)bb * COUT * PS + p0 + n;
#pragma unroll
  for (int mt = 0; mt < 4; ++mt)
#pragma unroll
    for (int r = 0; r < 8; ++r)
      ob[(size_t)(chbase + mt * 16 + r + 8 * hf) * PS] = acc[mt][r];
}

// =====================================================================
extern "C" void kernel_launch(void* const* d_in, const int* in_sizes, int n_in,
                              void* d_out, int out_size, void* d_ws,
                              size_t ws_size, hipStream_t stream) {
  const float* feature = (const float*)d_in[0];
  const float* extra   = (const float*)d_in[1];
  const float* w_ctx   = (const float*)d_in[2];
  const float* b_ctx   = (const float*)d_in[3];
  const float* w1      = (const float*)d_in[4];
  const float* b1      = (const float*)d_in[5];
  const float* w_s[3]  = {(const float*)d_in[6], (const float*)d_in[8],
                          (const float*)d_in[10]};
  const float* b_s[3]  = {(const float*)d_in[7], (const float*)d_in[9],
                          (const float*)d_in[11]};
  float* out = (float*)d_out;

  // workspace layout: x_ctx | S1 | S2   (~133 MB total; S1+S2 fit in 192MB L2)
  float* xctx = (float*)d_ws;
  float* S1 = xctx + (size_t)NB * Gc * PS;                 // +2,359,296
  float* S2 = S1 + (size_t)NBB * CIN * SPLANE;             // +15,415,296

  // 1) context conv (2 batches only; repeat is virtual)
  k_ctx_conv<<<(NB * TILES) / 8, 256, 0, stream>>>(feature, w_ctx, b_ctx, xctx);
  // 2) integral images of x and x^2
  k_int_rows<<<NBB * CIN, 256, 0, stream>>>(xctx, extra, S1, S2);
  k_int_cols<<<NBB * CIN, Ww, 0, stream>>>(S1, S2);
  // 3) first 1x1 conv -> out channels [0,68)
  k_conv1<<<(NBB * TILES) / 8, 256, 0, stream>>>(xctx, extra, w1, b1, out);
  // 4) three fused box-stat + conv stages -> channels [68,260)
  const int halves[3] = {2, 4, 8};
  for (int s = 0; s < 3; ++s) {
    k_scale_conv<<<(NBB * TILES) / 8, 256, 0, stream>>>(
        S1, S2, w_s[s], b_s[s], out, halves[s], C1OUT + 64 * s);
  }
}